// GCN_PECF_60988535603573
// MI455X (gfx1250) — compile-verified
//
#include <hip/hip_runtime.h>
#include <hip/hip_bf16.h>

typedef __attribute__((ext_vector_type(2))) float v2f;
typedef __attribute__((ext_vector_type(8))) float v8f;

#define DF 128   // node feature dim
#define RF 64    // edge feature dim

// ---------------------------------------------------------------------------
// degree / norm kernels
// ---------------------------------------------------------------------------
__global__ void init_deg_kernel(float* degS, float* degD, int n) {
    int i = blockIdx.x * blockDim.x + threadIdx.x;
    if (i < n) { degS[i] = 1.0f; degD[i] = 1.0f; }   // self-loop contributes 1
}

__global__ void deg_accum_kernel(const int* __restrict__ src,
                                 const int* __restrict__ dst,
                                 float* degS, float* degD, int e) {
    int i = blockIdx.x * blockDim.x + threadIdx.x;
    if (i < e) {
        atomicAdd(&degS[src[i]], 1.0f);
        atomicAdd(&degD[dst[i]], 1.0f);
    }
}

__global__ void rsqrt_kernel(float* degS, float* degD, int n) {
    int i = blockIdx.x * blockDim.x + threadIdx.x;
    if (i < n) {
        degS[i] = rsqrtf(degS[i]);
        degD[i] = rsqrtf(degD[i]);
    }
}

// hs = x * norm_src (row-wise); agg initialized to hs (self-loop edge).
// NOTE: agg may alias x (layer 2 in-place) — elementwise read-then-write is safe.
__global__ void scale_rows_kernel(const float* x, const float* __restrict__ norm,
                                  float* hs, float* agg, int total, int d) {
    int i = blockIdx.x * blockDim.x + threadIdx.x;
    if (i < total) {
        float v = x[i] * norm[i / d];
        hs[i]  = v;
        agg[i] = v;
    }
}

// agg[dst[e]] += hs[src[e]], 4 floats per thread
__global__ void scatter_add_kernel(const int* __restrict__ src,
                                   const int* __restrict__ dst,
                                   const float* __restrict__ hs,
                                   float* __restrict__ agg,
                                   int e, int dchunks, int d) {
    int t = blockIdx.x * blockDim.x + threadIdx.x;
    if (t >= e * dchunks) return;
    int ei = t / dchunks;
    int c  = (t - ei * dchunks) * 4;
    int s  = src[ei];
    int dd = dst[ei];
    const float4 v = *(const float4*)(hs + (size_t)s * d + c);
    float* ap = agg + (size_t)dd * d + c;
    atomicAdd(ap + 0, v.x);
    atomicAdd(ap + 1, v.y);
    atomicAdd(ap + 2, v.z);
    atomicAdd(ap + 3, v.w);
}

// self-loop rows of e_h: text is zero-filled -> e_h row = relb
__global__ void fill_bias_kernel(float* out, const float* __restrict__ bias,
                                 int total, int nc) {
    int i = blockIdx.x * blockDim.x + threadIdx.x;
    if (i < total) out[i] = bias[i % nc];
}

// ---------------------------------------------------------------------------
// WMMA GEMM: Out[M,NC] = act( (A[M,K] * rowscale) @ W[K,NC] + bias )
// One wave -> 16 rows x NC cols (NC/16 fp32 16x16 accumulators).
// fp32 path: V_WMMA_F32_16X16X4_F32, K stepped by 4.
//
// Fragment layouts (ISA 7.12.2):
//   A 16x4 : lane<16: M=lane,  v0=K0,v1=K1 ; lane>=16: M=lane-16, v0=K2,v1=K3
//   B 4x16 : lane<16: N=lane,  v0=K0,v1=K1 ; lane>=16: N=lane-16, v0=K2,v1=K3
//   C 16x16: vgpr r, lane<16 -> (M=r,   N=lane)
//            vgpr r, lane>=16-> (M=r+8, N=lane-16)
//
// W is staged into LDS pre-swizzled in fragment order so each lane fetches its
// B fragment with a single aligned ds_load_b64 (no cross-register shuffles):
//   ldsB[ ((ks*NT + t)*32 + lane)*2 + j ] = W[4*ks + 2*(lane>>4) + j][t*16 + (lane&15)]
// ---------------------------------------------------------------------------
template <int K, int NC>
__global__ __launch_bounds__(256)
void wmma_gemm_kernel(const float* __restrict__ A,
                      const float* __restrict__ W,
                      const float* __restrict__ bias,
                      const float* __restrict__ rowscale,
                      float* __restrict__ Out,
                      int M, int relu) {
    constexpr int NT  = NC / 16;
    constexpr int KS  = K / 4;
    __shared__ float ldsB[KS * NT * 64];   // == K*NC floats

    // stage W swizzled: one (ks, t, lane) slot = 2 consecutive floats
    for (int i = threadIdx.x; i < KS * NT * 32; i += 256) {
        const int ks  = i / (NT * 32);
        const int rem = i - ks * (NT * 32);
        const int t   = rem >> 5;
        const int l   = rem & 31;
        const int kb  = 4 * ks + 2 * (l >> 4);
        const int n   = t * 16 + (l & 15);
        ldsB[i * 2 + 0] = W[(kb + 0) * NC + n];
        ldsB[i * 2 + 1] = W[(kb + 1) * NC + n];
    }
    __syncthreads();

    const int wave = threadIdx.x >> 5;
    const int lane = threadIdx.x & 31;
    const int row0 = (blockIdx.x * 8 + wave) * 16;   // wave-uniform
    if (row0 >= M) return;                           // whole wave exits together

    const int half  = lane >> 4;       // 0 | 1
    const int lr    = lane & 15;
    const int myRow = row0 + lr;
    const bool rowOk = (myRow < M);
    // branchless A path: clamp row, zero contribution via scale factor
    const int   loadRow = rowOk ? myRow : (M - 1);
    const float rsBase  = (rowscale != nullptr) ? rowscale[loadRow] : 1.0f;
    const float rs      = rowOk ? rsBase : 0.0f;

    v8f acc[NT];
#pragma unroll
    for (int t = 0; t < NT; ++t)
#pragma unroll
        for (int i = 0; i < 8; ++i) acc[t][i] = 0.0f;

    const float* Arow = A + (size_t)loadRow * K + 2 * half;
    const float* Bw   = ldsB + lane * 2;

#pragma unroll 4
    for (int ks = 0; ks < KS; ++ks) {
        v2f a = *(const v2f*)(Arow + 4 * ks);   // unconditional, pipelinable
        a.x *= rs;
        a.y *= rs;
#pragma unroll
        for (int t = 0; t < NT; ++t) {
            v2f b = *(const v2f*)(Bw + (ks * NT + t) * 64);  // one ds_load_b64
            acc[t] = __builtin_amdgcn_wmma_f32_16x16x4_f32(
                false, a, false, b, (short)0, acc[t], false, false);
        }
    }

#pragma unroll
    for (int t = 0; t < NT; ++t) {
        const int n  = t * 16 + lr;
        const float bv = (bias != nullptr) ? bias[n] : 0.0f;
#pragma unroll
        for (int r = 0; r < 8; ++r) {
            const int m = row0 + r + 8 * half;
            if (m < M) {
                float v = acc[t][r] + bv;
                if (relu) v = fmaxf(v, 0.0f);
                Out[(size_t)m * NC + n] = v;
            }
        }
    }
}

// ---------------------------------------------------------------------------
extern "C" void kernel_launch(void* const* d_in, const int* in_sizes, int n_in,
                              void* d_out, int out_size, void* d_ws, size_t ws_size,
                              hipStream_t stream) {
    (void)n_in; (void)out_size; (void)ws_size;

    const int*   src    = (const int*)  d_in[0];
    const int*   dst    = (const int*)  d_in[1];
    const float* x      = (const float*)d_in[2];
    const float* text_h = (const float*)d_in[3];
    const float* W1     = (const float*)d_in[4];
    const float* b1     = (const float*)d_in[5];
    const float* W2     = (const float*)d_in[6];
    const float* b2     = (const float*)d_in[7];
    const float* relW   = (const float*)d_in[8];
    const float* relb   = (const float*)d_in[9];

    const int E = in_sizes[0];
    const int N = in_sizes[2] / DF;

    // workspace layout
    float* normS = (float*)d_ws;                 // [N]  out_deg -> rsqrt
    float* normD = normS + N;                    // [N]  in_deg  -> rsqrt
    float* bufA  = normD + N;                    // [N, DF]
    float* bufB  = bufA + (size_t)N * DF;        // [N, DF]

    float* out_h = (float*)d_out;                // [N, DF]
    float* out_e = out_h + (size_t)N * DF;       // [E+N, RF]

    const int TB = 256;
    const int ND = N * DF;

    // ---- degrees & norms (self loops included) ----
    init_deg_kernel<<<(N + TB - 1) / TB, TB, 0, stream>>>(normS, normD, N);
    deg_accum_kernel<<<(E + TB - 1) / TB, TB, 0, stream>>>(src, dst, normS, normD, E);
    rsqrt_kernel<<<(N + TB - 1) / TB, TB, 0, stream>>>(normS, normD, N);

    // ---- layer 1: hs=x*normS -> bufA, agg(init=hs) -> bufB ----
    scale_rows_kernel<<<(ND + TB - 1) / TB, TB, 0, stream>>>(x, normS, bufA, bufB, ND, DF);
    scatter_add_kernel<<<((E * (DF / 4)) + TB - 1) / TB, TB, 0, stream>>>(
        src, dst, bufA, bufB, E, DF / 4, DF);
    // h1 = relu((agg*normD) @ W1 + b1) -> bufA
    wmma_gemm_kernel<DF, DF><<<(N + 127) / 128, TB, 0, stream>>>(
        bufB, W1, b1, normD, bufA, N, 1);

    // ---- layer 2: hs2=h1*normS -> bufB, agg(init=hs2) -> bufA (in place) ----
    scale_rows_kernel<<<(ND + TB - 1) / TB, TB, 0, stream>>>(bufA, normS, bufB, bufA, ND, DF);
    scatter_add_kernel<<<((E * (DF / 4)) + TB - 1) / TB, TB, 0, stream>>>(
        src, dst, bufB, bufA, E, DF / 4, DF);
    // h = relu((agg*normD) @ W2 + b2) -> out_h   (final relu(relu(.)) idempotent)
    wmma_gemm_kernel<DF, DF><<<(N + 127) / 128, TB, 0, stream>>>(
        bufA, W2, b2, normD, out_h, N, 1);

    // ---- edge linear: e_h[0:E] = text_h @ relW + relb ----
    wmma_gemm_kernel<RF, RF><<<(E + 127) / 128, TB, 0, stream>>>(
        text_h, relW, relb, nullptr, out_e, E, 0);
    // e_h[E:E+N] = relb (zero text on self-loop edges)
    fill_bias_kernel<<<((N * RF) + TB - 1) / TB, TB, 0, stream>>>(
        out_e + (size_t)E * RF, relb, N * RF, RF);
}